// ST_BIFNeuron_MS_33002528703215
// MI455X (gfx1250) — compile-verified
//
#include <hip/hip_runtime.h>

namespace {

typedef float v4f __attribute__((ext_vector_type(4)));

constexpr int   kT      = 8;     // T_STEPS
constexpr float kQ      = 0.1f;  // q_threshold
constexpr float kInvQ   = 10.0f; // 1 / q_threshold
constexpr float kPosMax = 15.0f; // level - 1 (sym=False)

// One IF-neuron step. v/ta carried in registers; returns spike * q.
__device__ __forceinline__ float bif_step(float& v, float& ta, float xs) {
  v += xs;
  const bool  pos   = (v >= 1.0f) && (ta < kPosMax);
  const bool  neg   = (v <  0.0f) && (ta > 0.0f);
  const float spike = neg ? -1.0f : (pos ? 1.0f : 0.0f);
  v  -= spike;           // V_TH = 1.0; PREFIRE = 0 -> no prefire term
  ta += spike;
  return spike * kQ;
}

__device__ __forceinline__ void load_ba(const float* __restrict__ bias_alloc,
                                        float (&ba)[kT]) {
  // biasAllocator = cat([1 - sum(ba6), ba6])[:7]; zero beyond effect_T.
  const float a0 = bias_alloc[0], a1 = bias_alloc[1], a2 = bias_alloc[2],
              a3 = bias_alloc[3], a4 = bias_alloc[4], a5 = bias_alloc[5];
  ba[0] = 1.0f - (a0 + a1 + a2 + a3 + a4 + a5);
  ba[1] = a0; ba[2] = a1; ba[3] = a2; ba[4] = a3; ba[5] = a4; ba[6] = a5;
  ba[7] = 0.0f;  // effect_T == 7 < kT
}

// One block per (b,n) row; blockDim.x == C/4, so the channel index is
// structural (c = tid*4): no integer division/modulo anywhere.
// Each thread: 8 up-front strided b128 NT loads (max MLP), fully unrolled
// register scan over T, 8 b128 NT stores. Pure streaming, zero reuse.
__global__ void st_bif_row_vec4(const float* __restrict__ x,
                                const float* __restrict__ bias_alloc,
                                const float* __restrict__ bias_chan,
                                float* __restrict__ out,
                                unsigned long long stride,  // B*N*C elems
                                unsigned C) {
  const unsigned c = threadIdx.x * 4u;
  const unsigned long long base =
      (unsigned long long)blockIdx.x * C + c;

  float ba[kT];
  load_ba(bias_alloc, ba);  // uniform address -> scalar loads

  v4f bc = *reinterpret_cast<const v4f*>(bias_chan + c);  // 3KB, cache-hot
  bc *= kInvQ;

  v4f xv[kT];
#pragma unroll
  for (int t = 0; t < kT; ++t) {
    xv[t] = __builtin_nontemporal_load(
        reinterpret_cast<const v4f*>(x + (unsigned long long)t * stride + base));
  }

  float v0 = 0.5f, v1 = 0.5f, v2 = 0.5f, v3 = 0.5f;  // 0.5*V_TH init
  float c0 = 0.0f, c1 = 0.0f, c2 = 0.0f, c3 = 0.0f;  // Tacc

#pragma unroll
  for (int t = 0; t < kT; ++t) {
    const float w = ba[t];
    v4f o;
    o.x = bif_step(v0, c0, fmaf(w, bc.x, xv[t].x * kInvQ));
    o.y = bif_step(v1, c1, fmaf(w, bc.y, xv[t].y * kInvQ));
    o.z = bif_step(v2, c2, fmaf(w, bc.z, xv[t].z * kInvQ));
    o.w = bif_step(v3, c3, fmaf(w, bc.w, xv[t].w * kInvQ));
    __builtin_nontemporal_store(
        o, reinterpret_cast<v4f*>(out + (unsigned long long)t * stride + base));
  }
}

// Fallback for awkward C: one block per row, threads stride over channels.
// Still no division/modulo.
__global__ __launch_bounds__(256) void st_bif_row_scalar(
    const float* __restrict__ x,
    const float* __restrict__ bias_alloc,
    const float* __restrict__ bias_chan,
    float* __restrict__ out,
    unsigned long long stride, unsigned C) {
  float ba[kT];
  load_ba(bias_alloc, ba);

  const unsigned long long row =
      (unsigned long long)blockIdx.x * C;
  for (unsigned c = threadIdx.x; c < C; c += blockDim.x) {
    const unsigned long long base = row + c;
    const float bc = bias_chan[c] * kInvQ;
    float v = 0.5f, ta = 0.0f;
#pragma unroll
    for (int t = 0; t < kT; ++t) {
      const float xs = fmaf(ba[t], bc,
          __builtin_nontemporal_load(x + (unsigned long long)t * stride + base)
              * kInvQ);
      const float o = bif_step(v, ta, xs);
      __builtin_nontemporal_store(
          o, out + (unsigned long long)t * stride + base);
    }
  }
}

}  // namespace

extern "C" void kernel_launch(void* const* d_in, const int* in_sizes, int n_in,
                              void* d_out, int out_size, void* d_ws,
                              size_t ws_size, hipStream_t stream) {
  const float* x  = (const float*)d_in[0];  // [T*B, N, C] float32
  const float* ba = (const float*)d_in[1];  // [6] float32
  const float* bc = (const float*)d_in[2];  // [C] float32
  float* out = (float*)d_out;               // [T*B, N, C] float32

  const unsigned           C      = (unsigned)in_sizes[2];          // 768
  const unsigned long long stride =
      (unsigned long long)in_sizes[0] / kT;                         // B*N*C
  const unsigned           rows   = (unsigned)(stride / C);         // B*N

  const unsigned tpb = C / 4u;  // 192 for C=768 -> 6 wave32s per block
  if ((C % 4u == 0u) && tpb >= 32u && tpb <= 1024u && (tpb % 32u == 0u)) {
    st_bif_row_vec4<<<rows, tpb, 0, stream>>>(x, ba, bc, out, stride, C);
  } else {
    st_bif_row_scalar<<<rows, 256, 0, stream>>>(x, ba, bc, out, stride, C);
  }
}